// upperExtremityModel_22325240004628
// MI455X (gfx1250) — compile-verified
//
#include <hip/hip_runtime.h>
#include <cstdint>

// Problem constants (from the reference)
#define B_   2048
#define T_   512
#define J_   20
#define C_   40            // 2 muscles per joint
#define NB_  8             // batches per block
#define TC_  8             // time steps per LDS chunk
#define NTH  (NB_ * J_)    // 160 threads = 5 waves
#define NCH  (T_ / TC_)    // 64 chunks
#define BATCH_CHUNK_FLOATS (TC_ * C_)        // 320 floats = 1280 B per batch per chunk
#define CHUNK_FLOATS       (NB_ * BATCH_CHUNK_FLOATS)  // 2560 floats = 10 KB

#if defined(__HIP_DEVICE_COMPILE__) && __has_builtin(__builtin_amdgcn_global_load_async_to_lds_b128)
#define USE_ASYNC_BUILTIN 1
#else
#define USE_ASYNC_BUILTIN 0
#endif

#if USE_ASYNC_BUILTIN
typedef __attribute__((ext_vector_type(4))) int v4i_t;
typedef __attribute__((address_space(1))) v4i_t gv4i_t;   // global int4
typedef __attribute__((address_space(3))) v4i_t lv4i_t;   // LDS int4
#endif

__device__ __forceinline__ void async_copy_b128(const float* gsrc, float* ldst) {
#if USE_ASYNC_BUILTIN
  __builtin_amdgcn_global_load_async_to_lds_b128(
      (gv4i_t*)(void*)gsrc,
      (lv4i_t*)(void*)ldst,
      0, 0);
#elif defined(__HIP_DEVICE_COMPILE__)
  uint32_t loff = (uint32_t)(uintptr_t)ldst;
  asm volatile("global_load_async_to_lds_b128 %0, %1, off"
               :: "v"(loff), "v"(gsrc) : "memory");
#else
  *(float4*)ldst = *(const float4*)gsrc;  // host-pass stub
#endif
}

template <int N>
__device__ __forceinline__ void wait_asynccnt() {
#if defined(__HIP_DEVICE_COMPILE__)
#if __has_builtin(__builtin_amdgcn_s_wait_asynccnt)
  __builtin_amdgcn_s_wait_asynccnt(N);
#else
  if constexpr (N == 0) asm volatile("s_wait_asynccnt 0x0" ::: "memory");
  else                  asm volatile("s_wait_asynccnt 0x4" ::: "memory");
#endif
#endif
}

__global__ __launch_bounds__(NTH) void
muscle_joint_scan_kernel(const float* __restrict__ x,     // [B,T,2J]
                         const float* __restrict__ st0,   // [B,J,2]
                         float* __restrict__ out,         // [B,T,J]
                         float* __restrict__ stf)         // [B,J,2]
{
  __shared__ float lbuf[2][CHUNK_FLOATS];   // double-buffered chunk of x

  const int tid = threadIdx.x;
  const int b0  = blockIdx.x * NB_;
  const int bl  = tid / J_;           // batch-local index 0..7
  const int j   = tid - bl * J_;      // joint 0..19
  const int b   = b0 + bl;
  const int sidx = (b * J_ + j) * 2;  // state index

  float theta = st0[sidx + 0];
  float omega = st0[sidx + 1];

  // ---- Precomputed, strength-reduced copy addressing -------------------
  // Per batch row a chunk is TC_*C_*4 = 1280 contiguous bytes = 80 x b128.
  // 8 rows = 640 units; 160 threads -> 4 async b128 issues per thread.
  // Each per-thread global pointer advances by exactly one batch-chunk
  // (1280 B) per chunk; LDS targets are fixed, selected by buffer parity.
  const float* gp[4];
  float*       lp[4];
  #pragma unroll
  for (int k = 0; k < 4; ++k) {
    int i  = tid + NTH * k;
    int rb = i / 80;                  // batch row within block tile
    int w  = i - rb * 80;             // 16B unit within row
    gp[k] = x + (size_t)(b0 + rb) * (T_ * C_) + w * 4;
    lp[k] = &lbuf[0][rb * BATCH_CHUNK_FLOATS + w * 4];
  }

  auto issue = [&](int buf) {
    #pragma unroll
    for (int k = 0; k < 4; ++k) {
      async_copy_b128(gp[k], lp[k] + buf * CHUNK_FLOATS);
      gp[k] += BATCH_CHUNK_FLOATS;    // next chunk of this batch row
    }
  };

  const float* cb0 = &lbuf[0][bl * BATCH_CHUNK_FLOATS + 2 * j];
  float*       op  = out + (size_t)b * (T_ * J_) + j;   // advances by TC_*J_

  issue(0);

  for (int q = 0; q < NCH; ++q) {
    if (q + 1 < NCH) {
      issue((q + 1) & 1);            // prefetch next chunk
      wait_asynccnt<4>();            // current chunk's 4 issues complete
    } else {
      wait_asynccnt<0>();
    }
    __syncthreads();                 // all waves' async data visible

    const float* cb = cb0 + (q & 1) * CHUNK_FLOATS;

    #pragma unroll
    for (int tt = 0; tt < TC_; ++tt) {
      float2 a = *(const float2*)(cb + tt * C_);   // (a0, a1)
      // stretch = L0 + L1*a - arm*theta, arms = (-M, +M), M = 0.05
      float s0 = fmaf(0.006f, a.x, fmaf( 0.05f, theta, 0.06f));
      float s1 = fmaf(0.006f, a.y, fmaf(-0.05f, theta, 0.06f));
      // F = relu((K0 + K1*a) * stretch)
      float F0 = fmaxf(fmaf(2000.0f, a.x, 100.0f) * s0, 0.0f);
      float F1 = fmaxf(fmaf(2000.0f, a.y, 100.0f) * s1, 0.0f);
      float tau   = 0.05f * (F1 - F0);               // net torque
      float alpha = (tau - fmaf(5.0f, theta, 0.3f * omega)) * (1.0f / 0.004f);
      omega = fmaf(1.0f / 60.0f, alpha, omega);      // semi-implicit Euler
      theta = fmaf(1.0f / 60.0f, omega, theta);
      op[tt * J_] = theta;
    }
    op += TC_ * J_;
    __syncthreads();                 // buffer may be refilled next iteration
  }

  stf[sidx + 0] = theta;
  stf[sidx + 1] = omega;
}

extern "C" void kernel_launch(void* const* d_in, const int* in_sizes, int n_in,
                              void* d_out, int out_size, void* d_ws, size_t ws_size,
                              hipStream_t stream) {
  const float* x   = (const float*)d_in[0];       // [B,T,2J] f32
  const float* st0 = (const float*)d_in[1];       // [B,J,2]  f32
  float* out = (float*)d_out;                     // [B,T,J] then [B,J,2]
  float* stf = out + (size_t)B_ * T_ * J_;
  dim3 grid(B_ / NB_);     // 256 blocks
  dim3 block(NTH);         // 160 threads (5 wave32)
  hipLaunchKernelGGL(muscle_joint_scan_kernel, grid, block, 0, stream, x, st0, out, stf);
}